// GC_GCN_2293512536174
// MI455X (gfx1250) — compile-verified
//
#include <hip/hip_runtime.h>
#include <hip/hip_bf16.h>

// ---------------------------------------------------------------------------
// GCN layer + mean-pool + linear head for MI455X (gfx1250, wave32).
//   agg[dst] += feat[src]*outnorm[src]  (L2-resident atomic scatter, dominant)
//   h = relu((agg * innorm) @ W + b)    (f32 WMMA 16x16x4, fused pool epilogue)
//   out = (segment_mean(h)) @ W_head + b_head
// ---------------------------------------------------------------------------

typedef __attribute__((ext_vector_type(2))) float v2f;
typedef __attribute__((ext_vector_type(8))) float v8f;

#define N_NODES   50000
#define N_EDGES   800000
#define IN_DIM    128
#define OUT_DIM   128
#define N_CLASSES 10
#define N_GRAPHS  64

// Workspace layout (float offsets). Total ~26.03 MB.
#define WS_AGG     0
#define WS_DEGOUT  (N_NODES * IN_DIM)            // 6,400,000
#define WS_DEGIN   (WS_DEGOUT + N_NODES)
#define WS_POOL    (WS_DEGIN + N_NODES)
#define WS_CNT     (WS_POOL + N_GRAPHS * OUT_DIM)
#define WS_TOTAL   (WS_CNT + N_GRAPHS)           // 6,508,256 floats

// ---- zero the workspace (agg, degrees, pooled accumulators, counts) -------
__global__ void k_zero(float* __restrict__ ws) {
    int stride = gridDim.x * blockDim.x;
    for (int i = blockIdx.x * blockDim.x + threadIdx.x; i < WS_TOTAL; i += stride)
        ws[i] = 0.0f;
}

// ---- degree counting via f32 atomics (L2-resident, 50k-entry arrays) ------
__global__ void k_degree(const int* __restrict__ src, const int* __restrict__ dst,
                         float* __restrict__ deg_out, float* __restrict__ deg_in) {
    int e = blockIdx.x * blockDim.x + threadIdx.x;
    if (e < N_EDGES) {
        atomicAdd(&deg_out[src[e]], 1.0f);
        atomicAdd(&deg_in[dst[e]],  1.0f);
    }
}

// ---- deg -> rsqrt(max(deg,1)) in place; also count nodes per graph --------
__global__ void k_norm_cnt(float* __restrict__ deg_out, float* __restrict__ deg_in,
                           const int* __restrict__ gid, float* __restrict__ cnt) {
    int n = blockIdx.x * blockDim.x + threadIdx.x;
    if (n < N_NODES) {
        deg_out[n] = rsqrtf(fmaxf(deg_out[n], 1.0f));
        deg_in[n]  = rsqrtf(fmaxf(deg_in[n],  1.0f));
        atomicAdd(&cnt[gid[n]], 1.0f);
    }
}

// ---- edge scatter: one wave per edge, float4 gather, 4 atomics per lane ---
__global__ __launch_bounds__(256) void k_scatter(
        const float* __restrict__ feat, const int* __restrict__ src,
        const int* __restrict__ dst, const float* __restrict__ out_norm,
        float* __restrict__ agg) {
    int gtid = blockIdx.x * blockDim.x + threadIdx.x;
    int e    = gtid >> 5;          // wave-per-edge
    int lane = gtid & 31;          // 32 lanes x float4 = 128 dims
    if (e >= N_EDGES) return;
    int s = src[e];
    int d = dst[e];
    float nrm = out_norm[s];
    const float4 v = ((const float4*)(feat + (size_t)s * IN_DIM))[lane];
    float* p = agg + (size_t)d * IN_DIM + lane * 4;
    atomicAdd(p + 0, v.x * nrm);
    atomicAdd(p + 1, v.y * nrm);
    atomicAdd(p + 2, v.z * nrm);
    atomicAdd(p + 3, v.w * nrm);
}

// ---- fused GEMM (f32 WMMA 16x16x4) + bias + ReLU + graph mean-pool --------
// Grid: 3125 blocks (row tiles of 16; 50000 = 3125*16 exactly).
// Block: 256 threads = 8 waves; wave w owns column tile w (128 = 8*16).
// A fragment (16x4 f32, 2 VGPR): lane m = lane&15, K = k0 + 2*(lane>>4) + {0,1}
// B fragment (4x16 f32, 2 VGPR): lane n = lane&15, same K striping
// C/D (16x16 f32, 8 VGPR): vgpr r -> M = r + 8*(lane>>4), N = lane&15
__global__ __launch_bounds__(256) void k_gemm_relu_pool(
        const float* __restrict__ agg, const float* __restrict__ in_norm,
        const float* __restrict__ W,   const float* __restrict__ bias,
        const int* __restrict__ gid,   float* __restrict__ pooled) {
    const int wave = threadIdx.x >> 5;        // 0..7 -> column tile
    const int lane = threadIdx.x & 31;
    const int row0 = blockIdx.x * 16;
    const int col0 = wave * 16;
    const int hi   = lane >> 4;               // 0 or 1
    const int lo   = lane & 15;

    // Fold D_in^{-1/2} row scaling into the A-matrix load (row M = lo).
    const float        ascale = in_norm[row0 + lo];
    const float* __restrict__ arow = agg + (size_t)(row0 + lo) * IN_DIM;

    v8f c = {};
    #pragma unroll 4
    for (int k = 0; k < IN_DIM; k += 4) {
        const int ka = k + 2 * hi;
        v2f a, b;
        a.x = arow[ka]     * ascale;
        a.y = arow[ka + 1] * ascale;
        b.x = W[(size_t)ka       * OUT_DIM + col0 + lo];
        b.y = W[(size_t)(ka + 1) * OUT_DIM + col0 + lo];
        // D = A x B + C, f32 in / f32 accum: v_wmma_f32_16x16x4_f32
        c = __builtin_amdgcn_wmma_f32_16x16x4_f32(
                /*neg_a=*/false, a, /*neg_b=*/false, b,
                /*c_mod=*/(short)0, c, /*reuse_a=*/false, /*reuse_b=*/false);
    }

    // Epilogue: bias + ReLU + mean-pool accumulation. graph_ids is sorted, so
    // a 16-row tile usually maps to one graph: run-length-compress the 8
    // per-lane atomics down to ~1.
    const float bj = bias[col0 + lo];
    float acc   = 0.0f;
    int   gprev = gid[row0 + 8 * hi];
    #pragma unroll
    for (int r = 0; r < 8; ++r) {
        const int   m = r + 8 * hi;
        const float v = fmaxf(c[r] + bj, 0.0f);
        const int   g = gid[row0 + m];
        if (g == gprev) {
            acc += v;
        } else {
            atomicAdd(&pooled[gprev * OUT_DIM + col0 + lo], acc);
            gprev = g;
            acc   = v;
        }
    }
    atomicAdd(&pooled[gprev * OUT_DIM + col0 + lo], acc);
}

// ---- head: out[g,j] = (pooled[g,:]/cnt[g]) @ W_head[:,j] + b_head[j] ------
__global__ void k_head(const float* __restrict__ pooled, const float* __restrict__ cnt,
                       const float* __restrict__ Wh, const float* __restrict__ bh,
                       float* __restrict__ out) {
    int g = blockIdx.x;
    int j = threadIdx.x;
    if (j >= N_CLASSES) return;
    float inv = 1.0f / fmaxf(cnt[g], 1.0f);
    float acc = bh[j];
    for (int k = 0; k < OUT_DIM; ++k)
        acc = fmaf(pooled[g * OUT_DIM + k] * inv, Wh[k * N_CLASSES + j], acc);
    out[g * N_CLASSES + j] = acc;
}

extern "C" void kernel_launch(void* const* d_in, const int* in_sizes, int n_in,
                              void* d_out, int out_size, void* d_ws, size_t ws_size,
                              hipStream_t stream) {
    const float* feat = (const float*)d_in[0];   // [N, 128] f32
    const int*   src  = (const int*)d_in[1];     // [E] (harness delivers int32)
    const int*   dst  = (const int*)d_in[2];     // [E]
    const int*   gid  = (const int*)d_in[3];     // [N] sorted
    const float* W    = (const float*)d_in[4];   // [128, 128]
    const float* b    = (const float*)d_in[5];   // [128]
    const float* Wh   = (const float*)d_in[6];   // [128, 10]
    const float* bh   = (const float*)d_in[7];   // [10]
    float*       out  = (float*)d_out;           // [64, 10]

    float* ws      = (float*)d_ws;
    float* agg     = ws + WS_AGG;
    float* deg_out = ws + WS_DEGOUT;
    float* deg_in  = ws + WS_DEGIN;
    float* pooled  = ws + WS_POOL;
    float* cnt     = ws + WS_CNT;

    k_zero<<<4096, 256, 0, stream>>>(ws);
    k_degree<<<(N_EDGES + 255) / 256, 256, 0, stream>>>(src, dst, deg_out, deg_in);
    k_norm_cnt<<<(N_NODES + 255) / 256, 256, 0, stream>>>(deg_out, deg_in, gid, cnt);
    k_scatter<<<(N_EDGES * 32) / 256, 256, 0, stream>>>(feat, src, dst, deg_out, agg);
    k_gemm_relu_pool<<<N_NODES / 16, 256, 0, stream>>>(agg, deg_in, W, b, gid, pooled);
    k_head<<<N_GRAPHS, 32, 0, stream>>>(pooled, cnt, Wh, bh, out);
}